// ChamferDistance_10625749090594
// MI455X (gfx1250) — compile-verified
//
#include <hip/hip_runtime.h>
#include <math.h>

typedef float v2f __attribute__((ext_vector_type(2)));
typedef float v8f __attribute__((ext_vector_type(8)));

namespace {
constexpr int BATCH   = 16;
constexpr int NPTS    = 4096;   // points per cloud (both x and y)
constexpr int ROWS_PB = 128;    // x-rows per workgroup (8 waves * 16 rows)
constexpr int THREADS = 256;    // 8 wave32s
}

__global__ void chamfer_zero_out(float* out) {
  if (threadIdx.x == 0 && blockIdx.x == 0) out[0] = 0.0f;
}

// Computes  scale * sum_b sum_m  min_n || X[b,m] - Y[b,n] ||^2
// and atomically accumulates into out[0].
// Launch twice with (X,Y) and (Y,X) to get both chamfer directions.
__global__ __launch_bounds__(THREADS)
void chamfer_dir_wmma(const float* __restrict__ X,
                      const float* __restrict__ Y,
                      float* __restrict__ out,
                      float scale) {
  // Staged y[b]: one 16B record per point: {py.x, py.y, py.z, |py|^2}.
  // The |p|^2 slot doubles as the WMMA K=3 pad column; it multiplies A's
  // K=3 column which is identically 0, so it never affects the product.
  __shared__ float pts4[NPTS * 4];          // 64 KB

  const int b       = blockIdx.x;           // batch
  const int rowBase = blockIdx.y * ROWS_PB; // first x-row of this block
  const int tid     = threadIdx.x;
  const int lane    = tid & 31;
  const int wave    = tid >> 5;
  const int half    = lane >> 4;            // 0: K=0,1 lanes ; 1: K=2,3 lanes
  const int l16     = lane & 15;

  // ---- stage + pack y[b] into LDS, computing |y|^2 once per point
  {
    const float* yb = Y + (size_t)b * NPTS * 3;
    #pragma unroll 4
    for (int p = tid; p < NPTS; p += THREADS) {
      const float py0 = yb[3 * p + 0];
      const float py1 = yb[3 * p + 1];
      const float py2 = yb[3 * p + 2];
      float4 rec;
      rec.x = py0;
      rec.y = py1;
      rec.z = py2;
      rec.w = fmaf(py0, py0, fmaf(py1, py1, py2 * py2));
      *(float4*)(pts4 + 4 * p) = rec;
    }
  }
  __syncthreads();

  // ---- A matrix (16x4 f32) for this wave's 16 x-points, ISA VGPR layout:
  //   lanes 0-15 : VGPR0 = K0 (px), VGPR1 = K1 (py)   for M = l16
  //   lanes 16-31: VGPR0 = K2 (pz), VGPR1 = K3 (0)    for M = l16
  const float* xb = X + (size_t)b * NPTS * 3;
  const int myRow = rowBase + wave * 16 + l16;
  v2f a;
  a.x = xb[3 * myRow + 2 * half];                      // px (half0) / pz (half1)
  const float t1 = xb[3 * myRow + 1];                  // unconditional load
  a.y = (half == 0) ? t1 : 0.0f;                       // py / K=3 zero pad

  // |x[m]|^2 : half0 holds px^2+py^2, half1 holds pz^2 -> xor-16 add (setup only)
  float x2p = fmaf(a.x, a.x, a.y * a.y);
  x2p += __shfl_xor(x2p, 16);
  // per-lane x2 for its 8 D-rows: M = r + 8*half  (source lanes 0..15)
  float x2loc[8];
  #pragma unroll
  for (int r = 0; r < 8; ++r) x2loc[r] = __shfl(x2p, r + 8 * half);

  float rowmin[8];
  #pragma unroll
  for (int r = 0; r < 8; ++r) rowmin[r] = 3.0e38f;

  // ---- sweep all y-tiles of 16 points
  const int baddr = half * 2;                 // float offset within record
  #pragma unroll 4
  for (int ybase = 0; ybase < NPTS; ybase += 16) {
    const int n = ybase + l16;
    // B matrix (4x16 f32): half0 lanes -> (x,y), half1 lanes -> (z,|y|^2-pad)
    const v2f   bm  = *(const v2f*)(pts4 + 4 * n + baddr);   // one ds_load_b64
    const float y2n = pts4[4 * n + 3];                       // one ds_load_b32

    // cross[16x16] = A x B  (K=4; pad column multiplies A's zeros)
    v8f c = {};
    c = __builtin_amdgcn_wmma_f32_16x16x4_f32(
        /*neg_a=*/false, a, /*neg_b=*/false, bm,
        /*c_mod=*/(short)0, c, /*reuse_a=*/false, /*reuse_b=*/false);

    // d[m][n] = x2[m] + y2[n] - 2*cross ; track min_n of (y2[n] - 2*cross),
    // add the row-constant x2[m] once at the end.
    #pragma unroll
    for (int r = 0; r < 8; ++r) {
      const float dpart = fmaf(-2.0f, c[r], y2n);
      rowmin[r] = fminf(rowmin[r], dpart);
    }
  }

  // ---- reduce min over the 16 N-lanes within each half-wave
  #pragma unroll
  for (int r = 0; r < 8; ++r) {
    float v = rowmin[r];
    v = fminf(v, __shfl_xor(v, 1));
    v = fminf(v, __shfl_xor(v, 2));
    v = fminf(v, __shfl_xor(v, 4));
    v = fminf(v, __shfl_xor(v, 8));
    rowmin[r] = v + x2loc[r];        // full NN sq-dist for row M = r + 8*half
  }

  // ---- sum this wave's 16 row minima and accumulate (pre-scaled)
  float s = 0.0f;
  #pragma unroll
  for (int r = 0; r < 8; ++r) s += rowmin[r];
  s += __shfl_xor(s, 16);            // add the other half's 8 rows
  if (lane == 0) atomicAdd(out, s * scale);
}

extern "C" void kernel_launch(void* const* d_in, const int* in_sizes, int n_in,
                              void* d_out, int out_size, void* d_ws, size_t ws_size,
                              hipStream_t stream) {
  (void)in_sizes; (void)n_in; (void)d_ws; (void)ws_size; (void)out_size;
  const float* x = (const float*)d_in[0];
  const float* y = (const float*)d_in[1];
  float* out = (float*)d_out;

  chamfer_zero_out<<<dim3(1), dim3(64), 0, stream>>>(out);

  const dim3 grid(BATCH, NPTS / ROWS_PB);  // (16, 32)
  const dim3 block(THREADS);
  const float scale = 1.0f / (float)(BATCH * NPTS);
  // direction 1: mean_m min_n d(x_m, y_n)
  chamfer_dir_wmma<<<grid, block, 0, stream>>>(x, y, out, scale);
  // direction 2: mean_n min_m d(x_m, y_n)  (roles swapped, same kernel)
  chamfer_dir_wmma<<<grid, block, 0, stream>>>(y, x, out, scale);
}